// MultiLabelGHMLoss_1726576853378
// MI455X (gfx1250) — compile-verified
//
#include <hip/hip_runtime.h>
#include <stdint.h>

#define NUM_BINS    10
#define NUM_CLASSES 128
#define LAB_BINS    (3 * NUM_CLASSES)
#define ALPHA       0.999f

// Workspace layout (floats):
//  [0]         loss numerator sum
//  [1]         mask sum (broadcast over class dim)
//  [2..11]     GD histogram (10 bins)
//  [12..395]   label histogram (384 bins)
#define WS_FLOATS 396

// ---------------------------------------------------------------------------
// CDNA5 async global->LDS copy (gfx1250): GLOBAL_LOAD_ASYNC_TO_LDS_B128,
// GVS addressing (SGPR64 base + VGPR32 byte offset), tracked by ASYNCcnt.
// ---------------------------------------------------------------------------
__device__ __forceinline__ void async_ld_b128(uint32_t lds_off, uint64_t base, uint32_t byte_off) {
  asm volatile("global_load_async_to_lds_b128 %0, %1, %2"
               :: "v"(lds_off), "v"(byte_off), "s"(base)
               : "memory");
}
__device__ __forceinline__ void wait_async_le6() { asm volatile("s_wait_asynccnt 6" ::: "memory"); }
__device__ __forceinline__ void wait_async_le4() { asm volatile("s_wait_asynccnt 4" ::: "memory"); }
__device__ __forceinline__ void wait_async_le2() { asm volatile("s_wait_asynccnt 2" ::: "memory"); }
__device__ __forceinline__ void wait_async_le0() { asm volatile("s_wait_asynccnt 0" ::: "memory"); }

__global__ void ghm_init_kernel(float* __restrict__ ws) {
  int i = blockIdx.x * blockDim.x + threadIdx.x;
  if (i < WS_FLOATS) ws[i] = 0.0f;
}

#define PIPE_DEPTH 4
#define TILE_BYTES 4096u   // 256 threads * 16B per buffer

__launch_bounds__(256)
__global__ void ghm_main_kernel(const float* __restrict__ pred,
                                const float* __restrict__ targ,
                                const float* __restrict__ mask,
                                const float* __restrict__ gd_ema,
                                const float* __restrict__ lab_ema,
                                float* __restrict__ ws,
                                int R) {
  __shared__ float4 bufP[PIPE_DEPTH][256];   // pred tiles: 8 rows x 128 classes each
  __shared__ float4 bufT[PIPE_DEPTH][256];   // target tiles
  __shared__ float  s_inv_gd[NUM_BINS];
  __shared__ float  s_inv_lab[LAB_BINS];
  __shared__ float  s_hist_gd[NUM_BINS];
  __shared__ float  s_hist_lab[LAB_BINS];
  __shared__ float  s_red[256];

  const int tid  = (int)threadIdx.x;
  const int lane = tid & 31;           // lane within wave32
  const int wv   = tid >> 5;           // wave id 0..7 (one row per wave per iter)

  if (tid < NUM_BINS) { s_inv_gd[tid] = 1.0f / gd_ema[tid]; s_hist_gd[tid] = 0.0f; }
  for (int j = tid; j < LAB_BINS; j += 256) {
    s_inv_lab[j]  = 1.0f / lab_ema[j];
    s_hist_lab[j] = 0.0f;
  }
  __syncthreads();

  // Fixed class ownership: lane l handles classes c0..c0+3 for every row it touches.
  const int c0 = lane * 4;
  float invl[3][4];
#pragma unroll
  for (int k = 0; k < 4; ++k) {
#pragma unroll
    for (int s = 0; s < 3; ++s) invl[s][k] = s_inv_lab[(c0 + k) * 3 + s];
  }

  float loss_acc = 0.0f, m_acc = 0.0f;
  float gd_acc[NUM_BINS];
#pragma unroll
  for (int b = 0; b < NUM_BINS; ++b) gd_acc[b] = 0.0f;
  float lab_acc[4][3];
#pragma unroll
  for (int k = 0; k < 4; ++k)
#pragma unroll
    for (int s = 0; s < 3; ++s) lab_acc[k][s] = 0.0f;

  const uint64_t predB = (uint64_t)(uintptr_t)pred;
  const uint64_t targB = (uint64_t)(uintptr_t)targ;
  const uint32_t ldsP0 = (uint32_t)(uintptr_t)&bufP[0][tid];
  const uint32_t ldsT0 = (uint32_t)(uintptr_t)&bufT[0][tid];

  const int s1   = (int)gridDim.x * 8;          // row stride between a wave's iterations
  const int row0 = (int)blockIdx.x * 8 + wv;
  const uint32_t laneByte = (uint32_t)(c0 * 4);

  // Prologue: stage up to PIPE_DEPTH-1 = 3 tiles ahead.
  {
    const int r0 = row0, r1 = row0 + s1, r2 = row0 + 2 * s1;
    if (r0 < R) {
      uint32_t off = (uint32_t)r0 * (uint32_t)(NUM_CLASSES * 4) + laneByte;
      async_ld_b128(ldsP0, predB, off);
      async_ld_b128(ldsT0, targB, off);
    }
    if (r1 < R) {
      uint32_t off = (uint32_t)r1 * (uint32_t)(NUM_CLASSES * 4) + laneByte;
      async_ld_b128(ldsP0 + TILE_BYTES, predB, off);
      async_ld_b128(ldsT0 + TILE_BYTES, targB, off);
    }
    if (r2 < R) {
      uint32_t off = (uint32_t)r2 * (uint32_t)(NUM_CLASSES * 4) + laneByte;
      async_ld_b128(ldsP0 + 2u * TILE_BYTES, predB, off);
      async_ld_b128(ldsT0 + 2u * TILE_BYTES, targB, off);
    }
  }

  int buf = 0;
  for (int row = row0; row < R; row += s1) {
    const int n1 = row + s1, n2 = row + 2 * s1, n3 = row + 3 * s1;
    if (n3 < R) {
      // Issue 3 tiles ahead into the buffer freed last iteration.
      const uint32_t nb  = (uint32_t)((buf + 3) & (PIPE_DEPTH - 1)) * TILE_BYTES;
      const uint32_t off = (uint32_t)n3 * (uint32_t)(NUM_CLASSES * 4) + laneByte;
      async_ld_b128(ldsP0 + nb, predB, off);
      async_ld_b128(ldsT0 + nb, targB, off);
      wait_async_le6();       // 3 future tile-pairs may remain in flight
    } else if (n2 < R) {
      wait_async_le4();
    } else if (n1 < R) {
      wait_async_le2();
    } else {
      wait_async_le0();
    }

    const float m = mask[row];
    const float4 xp = bufP[buf][tid];
    const float4 xt = bufT[buf][tid];
    m_acc += 4.0f * m;

    const float xs[4]  = { xp.x, xp.y, xp.z, xp.w };
    const float tsv[4] = { xt.x, xt.y, xt.z, xt.w };
#pragma unroll
    for (int k = 0; k < 4; ++k) {
      const float x  = xs[k];
      const float tc = fminf(fmaxf(tsv[k], 0.0f), 1.0f);            // clip (LABEL_SMOOTHING=0)
      const float e  = __expf(-fabsf(x));                           // shared exp
      const float raw = fmaxf(x, 0.0f) - x * tc + __logf(1.0f + e); // stable BCE
      const float sig = ((x >= 0.0f) ? 1.0f : e) / (1.0f + e);      // sigmoid via same exp
      const float g   = fabsf(sig - tc);
      int bi  = (int)(g * (float)NUM_BINS); bi = bi > (NUM_BINS - 1) ? (NUM_BINS - 1) : bi;
      int sel = (int)(tc * 3.0f);           sel = sel > 2 ? 2 : sel;
      const float invgd = s_inv_gd[bi];
      const float invla = (sel == 0) ? invl[0][k] : ((sel == 1) ? invl[1][k] : invl[2][k]);
      loss_acc += raw * sqrtf(invgd * invla) * m;
#pragma unroll
      for (int b = 0; b < NUM_BINS; ++b) gd_acc[b] += (bi == b) ? m : 0.0f;
      lab_acc[k][0] += (sel == 0) ? m : 0.0f;
      lab_acc[k][1] += (sel == 1) ? m : 0.0f;
      lab_acc[k][2] += (sel == 2) ? m : 0.0f;
    }
    buf = (buf + 1) & (PIPE_DEPTH - 1);
  }

  // ---- block reductions ----
  s_red[tid] = loss_acc; __syncthreads();
  for (int s = 128; s > 0; s >>= 1) { if (tid < s) s_red[tid] += s_red[tid + s]; __syncthreads(); }
  if (tid == 0) atomicAdd(&ws[0], s_red[0]);
  __syncthreads();
  s_red[tid] = m_acc; __syncthreads();
  for (int s = 128; s > 0; s >>= 1) { if (tid < s) s_red[tid] += s_red[tid + s]; __syncthreads(); }
  if (tid == 0) atomicAdd(&ws[1], s_red[0]);

#pragma unroll
  for (int b = 0; b < NUM_BINS; ++b) atomicAdd(&s_hist_gd[b], gd_acc[b]);
#pragma unroll
  for (int k = 0; k < 4; ++k) {
    atomicAdd(&s_hist_lab[(c0 + k) * 3 + 0], lab_acc[k][0]);
    atomicAdd(&s_hist_lab[(c0 + k) * 3 + 1], lab_acc[k][1]);
    atomicAdd(&s_hist_lab[(c0 + k) * 3 + 2], lab_acc[k][2]);
  }
  __syncthreads();
  if (tid < NUM_BINS) atomicAdd(&ws[2 + tid], s_hist_gd[tid]);
  for (int j = tid; j < LAB_BINS; j += 256) atomicAdd(&ws[12 + j], s_hist_lab[j]);
}

__global__ void ghm_final_kernel(const float* __restrict__ gd_ema,
                                 const float* __restrict__ lab_ema,
                                 const float* __restrict__ ws,
                                 float* __restrict__ out) {
  __shared__ float sh[2];
  const int tid = (int)threadIdx.x;   // blockDim.x == LAB_BINS (384)
  if (tid < 2) sh[tid] = 0.0f;
  __syncthreads();

  const float h = ws[12 + tid];
  atomicAdd(&sh[0], h);
  __syncthreads();
  const float hn = h / fmaxf(sh[0], 1e-10f) * (float)LAB_BINS;
  const float e  = lab_ema[tid] * ALPHA + (1.0f - ALPHA) * hn;
  atomicAdd(&sh[1], e);
  __syncthreads();
  out[11 + tid] = e / fmaxf(sh[1], 1e-10f) * (float)LAB_BINS;

  if (tid == 0) {
    out[0] = ws[0] / fmaxf(ws[1], 1e-10f);
    float hs = 0.0f;
    for (int b = 0; b < NUM_BINS; ++b) hs += ws[2 + b];
    hs = fmaxf(hs, 1e-10f);
    float eb[NUM_BINS]; float es = 0.0f;
    for (int b = 0; b < NUM_BINS; ++b) {
      eb[b] = gd_ema[b] * ALPHA + (1.0f - ALPHA) * (ws[2 + b] / hs * (float)NUM_BINS);
      es += eb[b];
    }
    es = fmaxf(es, 1e-10f);
    for (int b = 0; b < NUM_BINS; ++b) out[1 + b] = eb[b] / es * (float)NUM_BINS;
  }
}

extern "C" void kernel_launch(void* const* d_in, const int* in_sizes, int n_in,
                              void* d_out, int out_size, void* d_ws, size_t ws_size,
                              hipStream_t stream) {
  (void)n_in; (void)out_size; (void)ws_size;
  const float* pred = (const float*)d_in[0];
  const float* targ = (const float*)d_in[1];
  const float* mask = (const float*)d_in[2];
  const float* gd   = (const float*)d_in[3];
  const float* lab  = (const float*)d_in[4];
  float* out = (float*)d_out;
  float* ws  = (float*)d_ws;

  const int R = in_sizes[2];   // B*T rows (mask element count)

  ghm_init_kernel<<<(WS_FLOATS + 255) / 256, 256, 0, stream>>>(ws);

  int grid = 1024;
  int maxg = (R + 7) / 8;
  if (grid > maxg) grid = maxg;
  if (grid < 1) grid = 1;
  ghm_main_kernel<<<grid, 256, 0, stream>>>(pred, targ, mask, gd, lab, ws, R);

  ghm_final_kernel<<<1, LAB_BINS, 0, stream>>>(gd, lab, ws, out);
}